// Voxels_40518721470753
// MI455X (gfx1250) — compile-verified
//
#include <hip/hip_runtime.h>

#define NB_VOXELS 128
#define N_POINTS  8388608

// Native clang vector type: required for __builtin_nontemporal_load/store.
typedef float v4f __attribute__((ext_vector_type(4)));

// sigmoid(x) = 1 / (1 + e^-x): v_exp_f32 + v_rcp_f32 (fast, fp32-accurate enough)
__device__ __forceinline__ float fast_sigmoid(float x) {
    return __frcp_rn(1.0f + __expf(-x));
}

__global__ __launch_bounds__(256) void voxel_gather_kernel(
    const float* __restrict__ X,        // [N,3] row-major
    const v4f*  __restrict__ vox,       // [128,128,128] of rgba
    float* __restrict__ rgb_out,        // [N,3] flat
    float* __restrict__ dens_out)       // [N]   flat
{
    // Each thread handles 4 consecutive points -> all memory ops are b128.
    const size_t t = (size_t)blockIdx.x * blockDim.x + threadIdx.x;   // 0 .. N/4-1

    const v4f* Xv = (const v4f*)X;
    const size_t xbase = 3 * t;

    // Prefetch the X data ~256 threads' worth ahead (streams through L2 once).
    __builtin_prefetch((const void*)(Xv + xbase + 3 * 256), 0, 0);

    // 48 contiguous bytes = 4 points of xyz, fully coalesced across the wave.
    v4f a = __builtin_nontemporal_load(Xv + xbase + 0);
    v4f b = __builtin_nontemporal_load(Xv + xbase + 1);
    v4f c = __builtin_nontemporal_load(Xv + xbase + 2);

    const float px[4] = {a.x, a.w, b.z, c.y};
    const float py[4] = {a.y, b.x, b.w, c.z};
    const float pz[4] = {a.z, b.y, c.x, c.w};

    float sr[4], sg[4], sb[4], dd[4];

    #pragma unroll
    for (int k = 0; k < 4; ++k) {
        const float x = px[k], y = py[k], z = pz[k];
        const bool cond = (fabsf(x) < 0.5f) & (fabsf(y) < 0.5f) & (fabsf(z) < 0.5f);

        // i = (floor(v*128 + 64) - 1) mod 128 ; '&127' == jnp.mod for pow2
        const int ix = (((int)floorf(fmaf(x, 128.0f, 64.0f))) - 1) & (NB_VOXELS - 1);
        const int iy = (((int)floorf(fmaf(y, 128.0f, 64.0f))) - 1) & (NB_VOXELS - 1);
        const int iz = (((int)floorf(fmaf(z, 128.0f, 64.0f))) - 1) & (NB_VOXELS - 1);

        // 16B-aligned gather; 33.5 MB table lives in the 192 MB L2 (RT policy).
        const size_t lin = ((size_t)ix * NB_VOXELS + (size_t)iy) * NB_VOXELS + (size_t)iz;
        const v4f g = vox[lin];

        const float r  = cond ? g.x : 0.0f;
        const float gg = cond ? g.y : 0.0f;
        const float bb = cond ? g.z : 0.0f;
        const float d  = cond ? g.w * 10.0f : 0.0f;

        sr[k] = fast_sigmoid(r);
        sg[k] = fast_sigmoid(gg);
        sb[k] = fast_sigmoid(bb);
        dd[k] = fmaxf(d, 0.0f);   // relu
    }

    // 12 contiguous floats of rgb per thread -> 3x b128 non-temporal stores.
    v4f* ro = (v4f*)(rgb_out + t * 12);
    v4f s0 = {sr[0], sg[0], sb[0], sr[1]};
    v4f s1 = {sg[1], sb[1], sr[2], sg[2]};
    v4f s2 = {sb[2], sr[3], sg[3], sb[3]};
    __builtin_nontemporal_store(s0, ro + 0);
    __builtin_nontemporal_store(s1, ro + 1);
    __builtin_nontemporal_store(s2, ro + 2);

    // 4 contiguous density floats -> 1x b128 non-temporal store.
    v4f dv = {dd[0], dd[1], dd[2], dd[3]};
    __builtin_nontemporal_store(dv, (v4f*)dens_out + t);
}

extern "C" void kernel_launch(void* const* d_in, const int* in_sizes, int n_in,
                              void* d_out, int out_size, void* d_ws, size_t ws_size,
                              hipStream_t stream) {
    (void)in_sizes; (void)n_in; (void)out_size; (void)d_ws; (void)ws_size;

    const float* X   = (const float*)d_in[0];               // N_POINTS*3 floats
    const v4f*   vox = (const v4f*)d_in[1];                 // 128^3 rgba
    float* rgb_out  = (float*)d_out;                        // first 3N floats
    float* dens_out = (float*)d_out + (size_t)N_POINTS * 3; // next N floats

    const int threads = 256;
    const int n_thr   = N_POINTS / 4;                       // 4 points per thread
    const int blocks  = n_thr / threads;                    // 8192

    voxel_gather_kernel<<<blocks, threads, 0, stream>>>(X, vox, rgb_out, dens_out);
}